// GCN_66297115181294
// MI455X (gfx1250) — compile-verified
//
#include <hip/hip_runtime.h>
#include <math.h>

// ---------------------------------------------------------------------------
// CDNA5 (gfx1250) implementation of the 3-branch GCN/GAT/GCN + readout +
// dense similarity products. All dense algebra runs through
// V_WMMA_F32_16X16X4_F32 (full f32 precision; the big products are
// output-bandwidth bound on MI455X so f32 WMMA is the right precision).
// ---------------------------------------------------------------------------

typedef float v2f __attribute__((ext_vector_type(2)));
typedef float v8f __attribute__((ext_vector_type(8)));

static constexpr int   Ff = 64;      // feature width
static constexpr int   Hh = 4;       // GAT heads
static constexpr float NEG_SLOPE = 0.2f;

// ---------------------------------------------------------------------------
// WMMA NT-GEMM: C(MxN) = A(MxK) @ B(NxK)^T [+ C] [+ bias(col)] [relu]
// One wave32 per 16x16 output tile; K stepped 4 at a time through
// v_wmma_f32_16x16x4_f32. Fragment layouts per CDNA5 ISA 7.12.2:
//   A: lane l<16 -> row m0+l, regs {K=kb, kb+1} with kb = k0 + 2*(l>=16)
//   B: lane l<16 -> col n0+l, regs {K=kb, kb+1}   (B is KxN = Bmat^T)
//   C/D: lane -> col n0+(l&15), reg r -> row m0 + r + 8*(l>=16)
// ---------------------------------------------------------------------------
__global__ void wmma_gemm_nt_kernel(const float* __restrict__ A, int lda,
                                    const float* __restrict__ B, int ldb,
                                    float* __restrict__ C, int ldc,
                                    int M, int N, int K,
                                    const float* __restrict__ bias,
                                    int accumulate, int relu)
{
    const int wave   = blockIdx.x * (blockDim.x >> 5) + (threadIdx.x >> 5);
    const int lane   = threadIdx.x & 31;
    const int tilesN = N >> 4;
    const int tilesM = M >> 4;
    const int tm = wave / tilesN;
    const int tn = wave - tm * tilesN;
    if (tm >= tilesM) return;                 // wave-uniform early exit

    const int m0   = tm << 4;
    const int n0   = tn << 4;
    const int half = lane >> 4;               // 0: K pair {0,1}; 1: {2,3}
    const int l15  = lane & 15;

    const float* __restrict__ Arow = A + (size_t)(m0 + l15) * lda;
    const float* __restrict__ Bcol = B + (size_t)(n0 + l15) * ldb;

    v8f acc;
    if (accumulate) {
#pragma unroll
        for (int r = 0; r < 8; ++r)
            acc[r] = C[(size_t)(m0 + r + 8 * half) * ldc + n0 + l15];
    } else {
#pragma unroll
        for (int r = 0; r < 8; ++r) acc[r] = 0.0f;
    }

    for (int k0 = 0; k0 < K; k0 += 4) {
        const int kb = k0 + 2 * half;
        v2f a, b;
        a[0] = Arow[kb];     a[1] = Arow[kb + 1];
        b[0] = Bcol[kb];     b[1] = Bcol[kb + 1];
        acc = __builtin_amdgcn_wmma_f32_16x16x4_f32(
                  false, a, false, b, (short)0, acc, false, false);
    }

    const float bv = bias ? bias[n0 + l15] : 0.0f;
#pragma unroll
    for (int r = 0; r < 8; ++r) {
        float v = acc[r] + bv;
        if (relu) v = fmaxf(v, 0.0f);
        C[(size_t)(m0 + r + 8 * half) * ldc + n0 + l15] = v;
    }
}

// ---------------------------------------------------------------------------
// Elementwise / graph-scatter kernels
// ---------------------------------------------------------------------------
__device__ __forceinline__ void atomicMaxF(float* addr, float val)
{
    int* ia  = (int*)addr;
    int  cur = __float_as_int(val);
    int  old = *ia;
    while (__int_as_float(old) < val) {
        int prev = atomicCAS(ia, old, cur);
        if (prev == old) break;
        old = prev;
    }
}

__global__ void fill_kernel(float* p, float v, int n)
{
    int i = blockIdx.x * blockDim.x + threadIdx.x;
    if (i < n) p[i] = v;
}

__global__ void deg_edge_kernel(const int* __restrict__ dst,
                                const float* __restrict__ ew,
                                float* deg, int E)
{
    int i = blockIdx.x * blockDim.x + threadIdx.x;
    if (i < E) atomicAdd(&deg[dst[i]], ew[i]);
}

__global__ void dinv_kernel(const float* __restrict__ deg, float* dinv, int N)
{
    int i = blockIdx.x * blockDim.x + threadIdx.x;
    if (i < N) { float d = deg[i]; dinv[i] = d > 0.0f ? rsqrtf(d) : 0.0f; }
}

// out[n,f] = h[n,f]*dinv[n]^2 + b[f]   (self-loop term + bias)
__global__ void gcn_self_bias_kernel(const float* __restrict__ h,
                                     const float* __restrict__ dinv,
                                     const float* __restrict__ b,
                                     float* out, int N)
{
    int idx = blockIdx.x * blockDim.x + threadIdx.x;
    if (idx >= N * Ff) return;
    int n = idx >> 6, f = idx & 63;
    float di = dinv[n];
    out[idx] = h[idx] * di * di + b[f];
}

__global__ void gcn_edge_kernel(const int* __restrict__ src,
                                const int* __restrict__ dst,
                                const float* __restrict__ ew,
                                const float* __restrict__ dinv,
                                const float* __restrict__ h,
                                float* out, int E)
{
    int idx = blockIdx.x * blockDim.x + threadIdx.x;
    if (idx >= E * Ff) return;
    int e = idx >> 6, f = idx & 63;
    int s = src[e], d = dst[e];
    float norm = dinv[s] * ew[e] * dinv[d];
    atomicAdd(&out[(size_t)d * Ff + f], h[(size_t)s * Ff + f] * norm);
}

__global__ void relu_kernel(float* p, int n)
{
    int i = blockIdx.x * blockDim.x + threadIdx.x;
    if (i < n) p[i] = fmaxf(p[i], 0.0f);
}

__global__ void gat_cnt_kernel(const int* __restrict__ dst,
                               const float* __restrict__ ew,
                               float* cnt, float* esum, int E)
{
    int i = blockIdx.x * blockDim.x + threadIdx.x;
    if (i < E) {
        int d = dst[i];
        atomicAdd(&cnt[d], 1.0f);
        atomicAdd(&esum[d], ew[i]);
    }
}

__global__ void loop_attr_kernel(const float* __restrict__ cnt,
                                 const float* __restrict__ esum,
                                 float* loopattr, int N)
{
    int i = blockIdx.x * blockDim.x + threadIdx.x;
    if (i < N) loopattr[i] = esum[i] / fmaxf(cnt[i], 1.0f);
}

// per (node, head): al_s / al_d dot products of the head slice with a_src/a_dst
__global__ void attn_node_kernel(const float* __restrict__ hg,
                                 const float* __restrict__ asrc,
                                 const float* __restrict__ adst,
                                 float* als, float* ald, int N)
{
    int idx = blockIdx.x * blockDim.x + threadIdx.x;
    if (idx >= N * Hh) return;
    int n = idx >> 2, h = idx & 3;
    const float* hv = hg + (size_t)n * (Hh * Ff) + h * Ff;
    const float* as = asrc + h * Ff;
    const float* ad = adst + h * Ff;
    float ss = 0.0f, sd = 0.0f;
#pragma unroll 8
    for (int f = 0; f < Ff; ++f) { ss += hv[f] * as[f]; sd += hv[f] * ad[f]; }
    als[idx] = ss;
    ald[idx] = sd;
}

// c[h] = sum_f We[h*F+f] * a_edge[h*F+f]   (ea @ We^T collapses to ea*c[h])
__global__ void cvec_kernel(const float* __restrict__ We,
                            const float* __restrict__ aedge, float* c)
{
    int h = threadIdx.x;
    if (h < Hh) {
        float s = 0.0f;
        for (int f = 0; f < Ff; ++f) s += We[h * Ff + f] * aedge[h * Ff + f];
        c[h] = s;
    }
}

__global__ void alpha_edge_kernel(const int* __restrict__ src,
                                  const int* __restrict__ dst,
                                  const float* __restrict__ ew,
                                  const float* __restrict__ als,
                                  const float* __restrict__ ald,
                                  const float* __restrict__ c,
                                  float* alpha, float* amax, int E)
{
    int idx = blockIdx.x * blockDim.x + threadIdx.x;
    if (idx >= E * Hh) return;
    int e = idx >> 2, h = idx & 3;
    int s = src[e], d = dst[e];
    float a = als[s * Hh + h] + ald[d * Hh + h] + ew[e] * c[h];
    a = a >= 0.0f ? a : NEG_SLOPE * a;
    alpha[idx] = a;
    atomicMaxF(&amax[d * Hh + h], a);
}

__global__ void alpha_loop_kernel(const float* __restrict__ loopattr,
                                  const float* __restrict__ als,
                                  const float* __restrict__ ald,
                                  const float* __restrict__ c,
                                  float* alpha, float* amax, int N, int E)
{
    int idx = blockIdx.x * blockDim.x + threadIdx.x;
    if (idx >= N * Hh) return;
    int n = idx >> 2, h = idx & 3;
    float a = als[idx] + ald[idx] + loopattr[n] * c[h];
    a = a >= 0.0f ? a : NEG_SLOPE * a;
    alpha[(size_t)(E + n) * Hh + h] = a;
    atomicMaxF(&amax[idx], a);
}

__global__ void ex_edge_kernel(const int* __restrict__ dst,
                               const float* __restrict__ amax,
                               float* alpha, float* denom, int E)
{
    int idx = blockIdx.x * blockDim.x + threadIdx.x;
    if (idx >= E * Hh) return;
    int e = idx >> 2, h = idx & 3;
    int d = dst[e];
    float v = __expf(alpha[idx] - amax[d * Hh + h]);
    alpha[idx] = v;
    atomicAdd(&denom[d * Hh + h], v);
}

__global__ void ex_loop_kernel(const float* __restrict__ amax,
                               float* alpha, float* denom, int N, int E)
{
    int idx = blockIdx.x * blockDim.x + threadIdx.x;
    if (idx >= N * Hh) return;
    int n = idx >> 2, h = idx & 3;
    size_t ai = (size_t)(E + n) * Hh + h;
    float v = __expf(alpha[ai] - amax[idx]);
    alpha[ai] = v;
    atomicAdd(&denom[idx], v);
}

__global__ void coef_edge_kernel(const int* __restrict__ dst,
                                 const float* __restrict__ denom,
                                 float* alpha, int E)
{
    int idx = blockIdx.x * blockDim.x + threadIdx.x;
    if (idx >= E * Hh) return;
    int e = idx >> 2, h = idx & 3;
    alpha[idx] /= denom[dst[e] * Hh + h];
}

__global__ void coef_loop_kernel(const float* __restrict__ denom,
                                 float* alpha, int N, int E)
{
    int idx = blockIdx.x * blockDim.x + threadIdx.x;
    if (idx >= N * Hh) return;
    alpha[(size_t)E * Hh + idx] /= denom[idx];
}

__global__ void gat_out_edge_kernel(const int* __restrict__ src,
                                    const int* __restrict__ dst,
                                    const float* __restrict__ hg,
                                    const float* __restrict__ coef,
                                    float* att, int E)
{
    int idx = blockIdx.x * blockDim.x + threadIdx.x;
    if (idx >= E * Ff) return;
    int e = idx >> 6, f = idx & 63;
    int s = src[e], d = dst[e];
    const float* hv = hg + (size_t)s * (Hh * Ff) + f;
    const float* cf = coef + (size_t)e * Hh;
    float v = hv[0] * cf[0] + hv[Ff] * cf[1] + hv[2 * Ff] * cf[2] + hv[3 * Ff] * cf[3];
    atomicAdd(&att[(size_t)d * Ff + f], 0.25f * v);   // mean over H=4 heads
}

__global__ void gat_out_loop_kernel(const float* __restrict__ hg,
                                    const float* __restrict__ coef,
                                    float* att, int N, int E)
{
    int idx = blockIdx.x * blockDim.x + threadIdx.x;
    if (idx >= N * Ff) return;
    int n = idx >> 6, f = idx & 63;
    const float* hv = hg + (size_t)n * (Hh * Ff) + f;
    const float* cf = coef + (size_t)(E + n) * Hh;
    float v = hv[0] * cf[0] + hv[Ff] * cf[1] + hv[2 * Ff] * cf[2] + hv[3 * Ff] * cf[3];
    atomicAdd(&att[idx], 0.25f * v);
}

__global__ void add_bias_relu_kernel(float* p, const float* __restrict__ b, int N)
{
    int idx = blockIdx.x * blockDim.x + threadIdx.x;
    if (idx >= N * Ff) return;
    p[idx] = fmaxf(p[idx] + b[idx & 63], 0.0f);
}

// ---------------------------------------------------------------------------
// Host-side orchestration
// ---------------------------------------------------------------------------
static inline int cdiv(int a, int b) { return (a + b - 1) / b; }

static void gemm_nt(hipStream_t s, const float* A, int lda, const float* B, int ldb,
                    float* C, int ldc, int M, int N, int K,
                    const float* bias, int accumulate, int relu)
{
    int tiles  = (M / 16) * (N / 16);
    int blocks = cdiv(tiles, 8);                // 8 wave32s per 256-thread block
    wmma_gemm_nt_kernel<<<blocks, 256, 0, s>>>(A, lda, B, ldb, C, ldc,
                                               M, N, K, bias, accumulate, relu);
}

struct Scratch {
    float *deg, *dinv, *cnt, *esum, *loopattr, *cvec;
    float *h1, *f1, *hg, *als, *ald, *alpha, *amax, *denom, *att, *h2, *f2;
};

static void run_branch(hipStream_t s, const Scratch& w,
                       const float* x, const int* src, const int* dst,
                       const float* ew, int N, int E,
                       const float* g1W, const float* g1b,
                       const float* gaW, const float* asrc, const float* adst,
                       const float* geW, const float* aedge, const float* gab,
                       const float* g2W, const float* g2b,
                       const float* cnnW, const float* cnnb,
                       float* fea)
{
    const int NT = 256;
    const int NF = N * Ff, EF = E * Ff, NH = N * Hh, EH = E * Hh;

    // ---- degrees / normalization (shared by both GCN layers) ----
    fill_kernel<<<cdiv(N, NT), NT, 0, s>>>(w.deg, 1.0f, N);   // self-loop weight
    deg_edge_kernel<<<cdiv(E, NT), NT, 0, s>>>(dst, ew, w.deg, E);
    dinv_kernel<<<cdiv(N, NT), NT, 0, s>>>(w.deg, w.dinv, N);

    // ---- GCN layer 1: f1 = relu(agg(x @ W1^T) + b1) ----
    gemm_nt(s, x, Ff, g1W, Ff, w.h1, Ff, N, Ff, Ff, nullptr, 0, 0);
    gcn_self_bias_kernel<<<cdiv(NF, NT), NT, 0, s>>>(w.h1, w.dinv, g1b, w.f1, N);
    gcn_edge_kernel<<<cdiv(EF, NT), NT, 0, s>>>(src, dst, ew, w.dinv, w.h1, w.f1, E);
    relu_kernel<<<cdiv(NF, NT), NT, 0, s>>>(w.f1, NF);

    // ---- GAT ----
    fill_kernel<<<cdiv(N, NT), NT, 0, s>>>(w.cnt, 0.0f, N);
    fill_kernel<<<cdiv(N, NT), NT, 0, s>>>(w.esum, 0.0f, N);
    gat_cnt_kernel<<<cdiv(E, NT), NT, 0, s>>>(dst, ew, w.cnt, w.esum, E);
    loop_attr_kernel<<<cdiv(N, NT), NT, 0, s>>>(w.cnt, w.esum, w.loopattr, N);

    gemm_nt(s, w.f1, Ff, gaW, Ff, w.hg, Hh * Ff, N, Hh * Ff, Ff, nullptr, 0, 0);
    attn_node_kernel<<<cdiv(NH, NT), NT, 0, s>>>(w.hg, asrc, adst, w.als, w.ald, N);
    cvec_kernel<<<1, 32, 0, s>>>(geW, aedge, w.cvec);

    fill_kernel<<<cdiv(NH, NT), NT, 0, s>>>(w.amax, -3.402823466e38f, NH);
    fill_kernel<<<cdiv(NH, NT), NT, 0, s>>>(w.denom, 0.0f, NH);
    alpha_edge_kernel<<<cdiv(EH, NT), NT, 0, s>>>(src, dst, ew, w.als, w.ald,
                                                  w.cvec, w.alpha, w.amax, E);
    alpha_loop_kernel<<<cdiv(NH, NT), NT, 0, s>>>(w.loopattr, w.als, w.ald,
                                                  w.cvec, w.alpha, w.amax, N, E);
    ex_edge_kernel<<<cdiv(EH, NT), NT, 0, s>>>(dst, w.amax, w.alpha, w.denom, E);
    ex_loop_kernel<<<cdiv(NH, NT), NT, 0, s>>>(w.amax, w.alpha, w.denom, N, E);
    coef_edge_kernel<<<cdiv(EH, NT), NT, 0, s>>>(dst, w.denom, w.alpha, E);
    coef_loop_kernel<<<cdiv(NH, NT), NT, 0, s>>>(w.denom, w.alpha, N, E);

    fill_kernel<<<cdiv(NF, NT), NT, 0, s>>>(w.att, 0.0f, NF);
    gat_out_edge_kernel<<<cdiv(EF, NT), NT, 0, s>>>(src, dst, w.hg, w.alpha, w.att, E);
    gat_out_loop_kernel<<<cdiv(NF, NT), NT, 0, s>>>(w.hg, w.alpha, w.att, N, E);
    add_bias_relu_kernel<<<cdiv(NF, NT), NT, 0, s>>>(w.att, gab, N);

    // ---- GCN layer 2: f2 = relu(agg(att @ W2^T) + b2) ----
    gemm_nt(s, w.att, Ff, g2W, Ff, w.h2, Ff, N, Ff, Ff, nullptr, 0, 0);
    gcn_self_bias_kernel<<<cdiv(NF, NT), NT, 0, s>>>(w.h2, w.dinv, g2b, w.f2, N);
    gcn_edge_kernel<<<cdiv(EF, NT), NT, 0, s>>>(src, dst, ew, w.dinv, w.h2, w.f2, E);
    relu_kernel<<<cdiv(NF, NT), NT, 0, s>>>(w.f2, NF);

    // ---- readout: fea = f1 @ W[:,0,:]^T + f2 @ W[:,1,:]^T + b ----
    gemm_nt(s, w.f1, Ff, cnnW,      2 * Ff, fea, Ff, N, Ff, Ff, cnnb, 0, 0);
    gemm_nt(s, w.f2, Ff, cnnW + Ff, 2 * Ff, fea, Ff, N, Ff, Ff, nullptr, 1, 0);
}

extern "C" void kernel_launch(void* const* d_in, const int* in_sizes, int n_in,
                              void* d_out, int out_size, void* d_ws, size_t ws_size,
                              hipStream_t stream)
{
    (void)in_sizes; (void)n_in; (void)out_size; (void)ws_size;
    constexpr int Nc = 8000, Nd = 6000, Ns = 6000;
    constexpr int Ec = 256000, Ed = 192000, Es = 192000;
    constexpr int MAXN = 8000, MAXE = 256000;

    // inputs
    const float* x_cir  = (const float*)d_in[0];
    const float* x_drug = (const float*)d_in[1];
    const float* x_dis  = (const float*)d_in[2];
    const int*   ce     = (const int*)d_in[3];     // (2, Ec)
    const float* cw     = (const float*)d_in[4];
    const int*   de     = (const int*)d_in[5];     // (2, Ed)
    const float* dw     = (const float*)d_in[6];
    const int*   se     = (const int*)d_in[7];     // (2, Es)
    const float* sw     = (const float*)d_in[8];
    const float* g_c1W  = (const float*)d_in[9];
    const float* g_c1b  = (const float*)d_in[10];
    const float* g_c2W  = (const float*)d_in[11];
    const float* g_c2b  = (const float*)d_in[12];
    const float* g_d1W  = (const float*)d_in[13];
    const float* g_d1b  = (const float*)d_in[14];
    const float* g_d2W  = (const float*)d_in[15];
    const float* g_d2b  = (const float*)d_in[16];
    const float* ga_cW  = (const float*)d_in[17];
    const float* ga_cs  = (const float*)d_in[18];
    const float* ga_cd  = (const float*)d_in[19];
    const float* ga_ceW = (const float*)d_in[20];
    const float* ga_cae = (const float*)d_in[21];
    const float* ga_cb  = (const float*)d_in[22];
    const float* ga_dW  = (const float*)d_in[23];
    const float* ga_ds  = (const float*)d_in[24];
    const float* ga_dd  = (const float*)d_in[25];
    const float* ga_deW = (const float*)d_in[26];
    const float* ga_dae = (const float*)d_in[27];
    const float* ga_db  = (const float*)d_in[28];
    const float* cnn_cW = (const float*)d_in[29];
    const float* cnn_cb = (const float*)d_in[30];
    const float* cnn_dW = (const float*)d_in[31];
    const float* cnn_db = (const float*)d_in[32];

    float* out = (float*)d_out;
    // output layout (flat, return order):
    //   [0]            cir_fea @ dis_fea^T   8000x6000
    //   [48,000,000]   cir_fea @ drug_fea^T  8000x6000
    //   [96,000,000]   drug_fea @ dis_fea^T  6000x6000
    //   [132,000,000]  cir_fea  8000x64
    //   [132,512,000]  drug_fea 6000x64
    //   [132,896,000]  dis_fea  6000x64
    float* cir_fea  = out + 132000000;
    float* drug_fea = out + 132512000;
    float* dis_fea  = out + 132896000;

    // workspace carve (float units), sized for the largest branch
    float* wsf = (float*)d_ws;
    size_t o = 0;
    auto alloc = [&](size_t n) { float* p = wsf + o; o += n; return p; };
    Scratch w;
    w.deg      = alloc(MAXN);
    w.dinv     = alloc(MAXN);
    w.cnt      = alloc(MAXN);
    w.esum     = alloc(MAXN);
    w.loopattr = alloc(MAXN);
    w.cvec     = alloc(8);
    w.h1       = alloc((size_t)MAXN * Ff);
    w.f1       = alloc((size_t)MAXN * Ff);
    w.hg       = alloc((size_t)MAXN * Hh * Ff);
    w.als      = alloc((size_t)MAXN * Hh);
    w.ald      = alloc((size_t)MAXN * Hh);
    w.alpha    = alloc((size_t)(MAXE + MAXN) * Hh);
    w.amax     = alloc((size_t)MAXN * Hh);
    w.denom    = alloc((size_t)MAXN * Hh);
    w.att      = alloc((size_t)MAXN * Ff);
    w.h2       = alloc((size_t)MAXN * Ff);
    w.f2       = alloc((size_t)MAXN * Ff);

    // branches (drug & dis reuse the "dis" weight set, per reference)
    run_branch(stream, w, x_cir,  ce, ce + Ec, cw, Nc, Ec,
               g_c1W, g_c1b, ga_cW, ga_cs, ga_cd, ga_ceW, ga_cae, ga_cb,
               g_c2W, g_c2b, cnn_cW, cnn_cb, cir_fea);
    run_branch(stream, w, x_drug, de, de + Ed, dw, Nd, Ed,
               g_d1W, g_d1b, ga_dW, ga_ds, ga_dd, ga_deW, ga_dae, ga_db,
               g_d2W, g_d2b, cnn_dW, cnn_db, drug_fea);
    run_branch(stream, w, x_dis,  se, se + Es, sw, Ns, Es,
               g_d1W, g_d1b, ga_dW, ga_ds, ga_dd, ga_deW, ga_dae, ga_db,
               g_d2W, g_d2b, cnn_dW, cnn_db, dis_fea);

    // final similarity products (output-bandwidth bound; f32 WMMA)
    gemm_nt(stream, cir_fea,  Ff, dis_fea,  Ff, out,             6000, Nc, Ns, Ff, nullptr, 0, 0);
    gemm_nt(stream, cir_fea,  Ff, drug_fea, Ff, out + 48000000,  6000, Nc, Nd, Ff, nullptr, 0, 0);
    gemm_nt(stream, drug_fea, Ff, dis_fea,  Ff, out + 96000000,  6000, Nd, Ns, Ff, nullptr, 0, 0);
}